// PillarEncoder_24326694765034
// MI455X (gfx1250) — compile-verified
//
#include <hip/hip_runtime.h>
#include <hip/hip_bf16.h>

typedef __attribute__((ext_vector_type(2))) float v2f;
typedef __attribute__((ext_vector_type(8))) float v8f;

#define BEV_W 512
#define BEV_H 512
#define SCALE_XY (BEV_W * BEV_H)   // 262144
#define C_OUT 64

// ---------------------------------------------------------------------------
// Kernel 1: zero a buffer with float4 stores (grid-stride)
// ---------------------------------------------------------------------------
__global__ void pe_zero_f4(float4* __restrict__ p, long n4) {
    long i = (long)blockIdx.x * blockDim.x + threadIdx.x;
    long stride = (long)gridDim.x * blockDim.x;
    float4 z = make_float4(0.f, 0.f, 0.f, 0.f);
    for (; i < n4; i += stride) p[i] = z;
}

// ---------------------------------------------------------------------------
// Kernel 2: scatter-mean accumulation (count + sum of xyz per pillar)
// ---------------------------------------------------------------------------
__global__ void pe_accum(const float* __restrict__ pts, int N,
                         float* __restrict__ cnt,
                         float* __restrict__ sx,
                         float* __restrict__ sy,
                         float* __restrict__ sz) {
    int i = blockIdx.x * blockDim.x + threadIdx.x;
    if (i >= N) return;
    const float* pt = pts + (long)i * 6;
    float x = pt[1], y = pt[2], z = pt[3];
    int b  = (int)pt[0];
    int bx = (int)floorf((x - (-51.2f)) / 0.2f);
    int by = (int)floorf((y - (-51.2f)) / 0.2f);
    int flat = b * SCALE_XY + by * BEV_W + bx;
    atomicAdd(&cnt[flat], 1.0f);
    atomicAdd(&sx[flat], x);
    atomicAdd(&sy[flat], y);
    atomicAdd(&sz[flat], z);
}

// ---------------------------------------------------------------------------
// Kernel 3: feature build + WMMA f32 GEMM (K padded 10->12) + BN + ReLU +
//           scatter-max into (B, C, H, W) BEV grid.
// One wave32 handles 16 points; 4 WMMA accumulators of 16x16 cover C=64.
// ---------------------------------------------------------------------------
__global__ void __launch_bounds__(256)
pe_wmma_scatter(const float* __restrict__ pts, int N,
                const float* __restrict__ Wm,
                const float* __restrict__ gamma,
                const float* __restrict__ beta,
                const float* __restrict__ rmean,
                const float* __restrict__ rvar,
                const float* __restrict__ cnt,
                const float* __restrict__ sx,
                const float* __restrict__ sy,
                const float* __restrict__ sz,
                float* __restrict__ out, int ntiles) {
    const int lane = threadIdx.x & 31;
    const int half = lane >> 4;        // 0: lanes 0-15, 1: lanes 16-31
    const int l15  = lane & 15;
    const int wave   = blockIdx.x * (blockDim.x >> 5) + (threadIdx.x >> 5);
    const int nwaves = gridDim.x * (blockDim.x >> 5);

    // --- B-matrix fragments: W is (10,64) row-major, padded to K=12 -------
    // f32 4x16 B layout: VGPR0 = row K (lanes 0-15) / K+2 (lanes 16-31),
    // VGPR1 = row K+1 / K+3.  (Mirrors A's 16x4 layout.)
    v2f bf[4][3];
#pragma unroll
    for (int n = 0; n < 4; ++n) {
#pragma unroll
        for (int f = 0; f < 3; ++f) {
            int col = n * 16 + l15;
            int k0  = f * 4 + 2 * half;
            bf[n][f].x = (k0     < 10) ? Wm[k0 * C_OUT + col]       : 0.f;
            bf[n][f].y = (k0 + 1 < 10) ? Wm[(k0 + 1) * C_OUT + col] : 0.f;
        }
    }

    // --- BN scale/bias: channel is constant per lane within a tile -------
    float scl[4], bia[4];
#pragma unroll
    for (int n = 0; n < 4; ++n) {
        int ch   = n * 16 + l15;
        float s  = gamma[ch] * rsqrtf(rvar[ch] + 1e-3f);
        scl[n]   = s;
        bia[n]   = beta[ch] - rmean[ch] * s;
    }

    for (int tile = wave; tile < ntiles; tile += nwaves) {
        int p = tile * 16 + l15;      // lanes 16-31 mirror lanes 0-15
        // named scalars (no dynamically-indexed array -> no cndmask chains)
        float f0 = 0.f, f1 = 0.f, f2 = 0.f, f3 = 0.f, f4 = 0.f;
        float f5 = 0.f, f6 = 0.f, f7 = 0.f, f8 = 0.f, f9 = 0.f;
        int base = -1;

        if (p < N) {
            const float* pt = pts + (long)p * 6;
            float x = pt[1], y = pt[2], z = pt[3];
            int b  = (int)pt[0];
            int bx = (int)floorf((x - (-51.2f)) / 0.2f);
            int by = (int)floorf((y - (-51.2f)) / 0.2f);
            int flat = b * SCALE_XY + by * BEV_W + bx;
            float inv = 1.0f / fmaxf(cnt[flat], 1.0f);
            f0 = x;  f1 = y;  f2 = z;  f3 = pt[4];  f4 = pt[5];
            f5 = x - sx[flat] * inv;
            f6 = y - sy[flat] * inv;
            f7 = z - sz[flat] * inv;
            f8 = x - (bx * 0.2f + (-51.1f));   // voxel/2 + pc_range = -51.1
            f9 = y - (by * 0.2f + (-51.1f));
            base = b * (C_OUT * SCALE_XY) + by * BEV_W + bx;
        }

        // --- A fragments (16x4 f32): VGPR0 = K/K+2, VGPR1 = K+1/K+3 ------
        // half==0 lanes carry K, K+1; half==1 lanes carry K+2, K+3.
        v2f a0, a1, a2;
        a0.x = half ? f2 : f0;  a0.y = half ? f3 : f1;
        a1.x = half ? f6 : f4;  a1.y = half ? f7 : f5;
        a2.x = half ? 0.f : f8; a2.y = half ? 0.f : f9;  // K=10,11 are pad

        // --- pillar base of point m = r + 8*half, hoisted out of n-loop --
        int bm[8];
#pragma unroll
        for (int r = 0; r < 8; ++r)
            bm[r] = __shfl(base, r + 8 * half, 32);

#pragma unroll
        for (int n = 0; n < 4; ++n) {
            v8f acc = {};
            acc = __builtin_amdgcn_wmma_f32_16x16x4_f32(
                false, a0, false, bf[n][0], (short)0, acc, false, false);
            acc = __builtin_amdgcn_wmma_f32_16x16x4_f32(
                false, a1, false, bf[n][1], (short)0, acc, false, false);
            acc = __builtin_amdgcn_wmma_f32_16x16x4_f32(
                false, a2, false, bf[n][2], (short)0, acc, false, false);

            int   ch = n * 16 + l15;
            float s  = scl[n], bb = bia[n];
            float* outc = out + (long)ch * SCALE_XY;
#pragma unroll
            for (int r = 0; r < 8; ++r) {
                // C/D layout: VGPR r holds M = r + 8*half, N = lane&15
                float v = fmaxf(acc[r] * s + bb, 0.f);
                if (bm[r] >= 0 && v > 0.f) {
                    // post-ReLU values are >= 0: uint atomicMax == float max
                    atomicMax((unsigned int*)(outc + bm[r]), __float_as_uint(v));
                }
            }
        }
    }
}

// ---------------------------------------------------------------------------
extern "C" void kernel_launch(void* const* d_in, const int* in_sizes, int n_in,
                              void* d_out, int out_size, void* d_ws, size_t ws_size,
                              hipStream_t stream) {
    const float* pts   = (const float*)d_in[0];
    const float* Wm    = (const float*)d_in[1];
    const float* gamma = (const float*)d_in[2];
    const float* beta  = (const float*)d_in[3];
    const float* rmean = (const float*)d_in[4];
    const float* rvar  = (const float*)d_in[5];
    float* out = (float*)d_out;

    const int  N       = in_sizes[0] / 6;
    const int  B       = out_size / (C_OUT * SCALE_XY);
    const long num_seg = (long)B * SCALE_XY;

    float* cnt = (float*)d_ws;          // num_seg floats
    float* sx  = cnt + num_seg;
    float* sy  = sx + num_seg;
    float* sz  = sy + num_seg;

    // 1) zero output grid (256 MB) and scatter-mean workspace (16 MB)
    pe_zero_f4<<<2048, 256, 0, stream>>>((float4*)out, (long)out_size / 4);
    pe_zero_f4<<<512, 256, 0, stream>>>((float4*)cnt, num_seg);  // 4*num_seg/4

    // 2) per-pillar count + xyz sums
    pe_accum<<<(N + 255) / 256, 256, 0, stream>>>(pts, N, cnt, sx, sy, sz);

    // 3) WMMA pointnet + BN + ReLU + scatter-max
    const int ntiles  = (N + 15) / 16;
    const int nblocks = (ntiles + 7) / 8;   // 8 wave32 per 256-thread block
    pe_wmma_scatter<<<nblocks, 256, 0, stream>>>(
        pts, N, Wm, gamma, beta, rmean, rvar, cnt, sx, sy, sz, out, ntiles);
}